// MFModel_10058813407397
// MI455X (gfx1250) — compile-verified
//
#include <hip/hip_runtime.h>
#include <hip/hip_bf16.h>

typedef float v2f __attribute__((ext_vector_type(2)));
typedef float v8f __attribute__((ext_vector_type(8)));

#define HIDDEN 64
#define WAVES_PER_BLOCK 4
#define ROWS_PER_WAVE 16

// One wave32 computes 16 sigmoid(dot(u_i, v_i)) outputs via
// diag( U(16x64) x V^T(64x16) ) using 16 chained V_WMMA_F32_16X16X4_F32.
__global__ __launch_bounds__(WAVES_PER_BLOCK * 32)
void mf_dot_sigmoid_wmma(const int* __restrict__ x,
                         const float* __restrict__ user_table,
                         const float* __restrict__ item_table,
                         float* __restrict__ out,
                         int n_out)
{
    const int lane = threadIdx.x & 31;
    const int wave = threadIdx.x >> 5;
    const int tile = blockIdx.x * WAVES_PER_BLOCK + wave;
    const int ntiles = (n_out + ROWS_PER_WAVE - 1) / ROWS_PER_WAVE;
    if (tile >= ntiles) return;                 // uniform per wave: EXEC stays all-1s

    const int base = tile * ROWS_PER_WAVE;

    // A-matrix f32 16x4 layout: lanes 0-15 carry M=0..15 with K={0,1} in
    // VGPR{0,1}; lanes 16-31 carry the same M rows with K={2,3}.
    // B 4x16 mirrors this with N in place of M. So every lane loads a float2
    // at row[lane&15] + 4*chunk + 2*(lane>>4) for BOTH operands.
    const int row  = lane & 15;
    const int koff = (lane >> 4) << 1;

    int pair = base + row;
    if (pair >= n_out) pair = n_out - 1;        // clamp (no divergence created)

    const int uidx = x[2 * pair + 0];
    const int vidx = x[2 * pair + 1];

    const float* up = user_table + (size_t)uidx * HIDDEN + koff;
    const float* vp = item_table + (size_t)vidx * HIDDEN + koff;

    // Hoist all 32 float2 gathers ahead of the WMMA chain so vmem issues batch
    // and the 16-deep accumulate chain hides under load latency.
    v2f ua[16], va[16];
#pragma unroll
    for (int c = 0; c < 16; ++c) {
        ua[c] = *(const v2f*)(up + 4 * c);
        va[c] = *(const v2f*)(vp + 4 * c);
    }

    v8f acc = {};
#pragma unroll
    for (int c = 0; c < 16; ++c) {
        acc = __builtin_amdgcn_wmma_f32_16x16x4_f32(
            /*neg_a=*/false, ua[c], /*neg_b=*/false, va[c],
            /*c_mod=*/(short)0, acc, /*reuse_a=*/false, /*reuse_b=*/false);
    }

    // Extract diag(D). C/D layout: VGPR r holds M=r (lanes 0-15, N=lane) and
    // M=r+8 (lanes 16-31, N=lane-16).
    //   D[m][m], m<8 :  vgpr m,    lane m
    //   D[m][m], m>=8:  vgpr m-8,  lane 16+m
    // Lane m wants component (m&7); source lane = m for m<8, 16+m for m>=8.
    // One bpermute per component serves both halves: 8 ds ops total.
    const int srcLane = (lane & 8) ? (16 + lane) : lane;
    float diag = 0.0f;
#pragma unroll
    for (int r = 0; r < 8; ++r) {
        float d = __shfl(acc[r], srcLane, 32);
        diag = ((lane & 7) == r) ? d : diag;
    }

    if (lane < ROWS_PER_WAVE && (base + lane) < n_out) {
        out[base + lane] = 1.0f / (1.0f + __expf(-diag));
    }
}

extern "C" void kernel_launch(void* const* d_in, const int* in_sizes, int n_in,
                              void* d_out, int out_size, void* d_ws, size_t ws_size,
                              hipStream_t stream) {
    const int*   x          = (const int*)d_in[0];
    const float* user_table = (const float*)d_in[1];
    const float* item_table = (const float*)d_in[2];
    float*       out        = (float*)d_out;

    const int rows_per_block = WAVES_PER_BLOCK * ROWS_PER_WAVE; // 64
    const int nblocks = (out_size + rows_per_block - 1) / rows_per_block; // 256 for BATCH=16384

    mf_dot_sigmoid_wmma<<<nblocks, WAVES_PER_BLOCK * 32, 0, stream>>>(
        x, user_table, item_table, out, out_size);
}